// GCNN3L_11785390260548
// MI455X (gfx1250) — compile-verified
//
#include <hip/hip_runtime.h>

// GCN, 3 layers of 64->64 + final 64->8.
// fp32 end-to-end (matches reference numerics); GEMMs on the CDNA5 WMMA
// pipe via V_WMMA_F32_16X16X4_F32; message passing via native f32 atomics.
// Assumes N % 16 == 0 (N = 100000 in this problem).

typedef __attribute__((ext_vector_type(2))) float v2f;
typedef __attribute__((ext_vector_type(8))) float v8f;

#define FEAT 64

__device__ __forceinline__ void atomAddF(float* p, float v) {
    // relaxed, agent scope -> global_atomic_add_f32 (no CAS loop)
    __hip_atomic_fetch_add(p, v, __ATOMIC_RELAXED, __HIP_MEMORY_SCOPE_AGENT);
}

// ---------------- degree / symmetric norm ----------------
__global__ void k_init_deg(float* __restrict__ deg, int n) {
    int i = blockIdx.x * blockDim.x + threadIdx.x;
    if (i < n) deg[i] = 1.0f;  // self loop
}

__global__ void k_accum_deg(const long long* __restrict__ dst, float* __restrict__ deg, int e) {
    int i = blockIdx.x * blockDim.x + threadIdx.x;
    if (i < e) atomAddF(&deg[(int)dst[i]], 1.0f);
}

__global__ void k_rsqrt_inplace(float* __restrict__ deg, int n) {
    int i = blockIdx.x * blockDim.x + threadIdx.x;
    if (i < n) deg[i] = rsqrtf(deg[i]);  // deg >= 1 always (self loops)
}

// ---------------- fp32 WMMA GEMM: T = act(H) @ W (+ bias) ----------------
// One wave computes one 16x16 output tile; K = 64 = 16 chained k=4 WMMAs.
// A layout (16x4 f32): lanes 0-15 hold K={0,1}, lanes 16-31 hold K={2,3}, M = lane&15.
// B layout (4x16 f32): mirrored — lanes 0-15 hold K={0,1}, 16-31 K={2,3}, N = lane&15.
// C/D layout (16x16 f32): VGPR r, lanes 0-15 -> (M=r, N=lane); lanes 16-31 -> (M=r+8, N=lane-16).
template <bool RELU, int FOUT, bool ADD_BIAS>
__global__ void k_gemm_wmma(const float* __restrict__ H, const float* __restrict__ W,
                            const float* __restrict__ bias, float* __restrict__ T) {
    const int lane    = threadIdx.x & 31;
    const int wave    = threadIdx.x >> 5;
    const int rowBase = blockIdx.x * 16;
    const int colBase = wave * 16;        // 4 waves cover FOUT=64; 1 wave for FOUT=8
    const int mh      = lane & 15;
    const int kk      = (lane >> 4) * 2;  // which K-pair this half-wave owns
    const int col     = colBase + mh;
    const float* hRow = H + (size_t)(rowBase + mh) * FEAT;

    v8f acc = {0.f, 0.f, 0.f, 0.f, 0.f, 0.f, 0.f, 0.f};

#pragma unroll
    for (int k = 0; k < FEAT / 4; ++k) {
        const int kb = k * 4 + kk;
        float a0 = hRow[kb];
        float a1 = hRow[kb + 1];
        if (RELU) { a0 = fmaxf(a0, 0.0f); a1 = fmaxf(a1, 0.0f); }
        float b0, b1;
        if (FOUT >= 16) {
            b0 = W[kb * FOUT + col];
            b1 = W[(kb + 1) * FOUT + col];
        } else {  // pad narrow weight (64x8) to 16 cols with zeros
            b0 = (col < FOUT) ? W[kb * FOUT + col] : 0.0f;
            b1 = (col < FOUT) ? W[(kb + 1) * FOUT + col] : 0.0f;
        }
        v2f a = {a0, a1};
        v2f b = {b0, b1};
        acc = __builtin_amdgcn_wmma_f32_16x16x4_f32(
            /*neg_a=*/false, a, /*neg_b=*/false, b,
            /*c_mod=*/(short)0, acc, /*reuse_a=*/false, /*reuse_b=*/false);
    }

    float bb = 0.0f;
    if (ADD_BIAS && (FOUT >= 16 || col < FOUT)) bb = bias[col];
    const int rOff = (lane >> 4) * 8;
#pragma unroll
    for (int r = 0; r < 8; ++r) {
        const int row = rowBase + rOff + r;
        const float v = acc[r] + (ADD_BIAS ? bb : 0.0f);
        if (FOUT >= 16) {
            T[(size_t)row * FOUT + col] = v;
        } else if (col < FOUT) {
            T[(size_t)row * FOUT + col] = v;
        }
    }
}

// ---------------- agg init: bias + self-loop contribution ----------------
// agg[i][:] = b[:] + T[i][:] * dis[i]^2   (plain stores, before edge atomics)
__global__ void k_init_agg(const float* __restrict__ T, const float* __restrict__ dis,
                           const float* __restrict__ bias, float* __restrict__ AGG, int n) {
    int t = blockIdx.x * blockDim.x + threadIdx.x;  // n*16 threads, float4 each
    if (t >= n * 16) return;
    const int node = t >> 4;
    const int f4   = t & 15;
    float s = dis[node];
    s = s * s;
    const float4 v = ((const float4*)(T + (size_t)node * FEAT))[f4];
    const int f = f4 * 4;
    float4 o;
    o.x = bias[f + 0] + v.x * s;
    o.y = bias[f + 1] + v.y * s;
    o.z = bias[f + 2] + v.z * s;
    o.w = bias[f + 3] + v.w * s;
    ((float4*)(AGG + (size_t)node * FEAT))[f4] = o;
}

// ---------------- edge scatter: agg[dst] += T[src] * dis[src]*dis[dst] ----------------
__global__ void k_scatter(const float* __restrict__ T, const float* __restrict__ dis,
                          const long long* __restrict__ src, const long long* __restrict__ dst,
                          float* __restrict__ AGG, int e) {
    long long t = (long long)blockIdx.x * blockDim.x + threadIdx.x;  // e*16 threads
    if (t >= (long long)e * 16) return;
    const int edge = (int)(t >> 4);
    const int f4   = (int)(t & 15);
    const int s    = (int)src[edge];
    const int d    = (int)dst[edge];
    const float w  = dis[s] * dis[d];
    const float4 v = ((const float4*)(T + (size_t)s * FEAT))[f4];
    float* base = AGG + (size_t)d * FEAT + (size_t)f4 * 4;
    atomAddF(base + 0, v.x * w);
    atomAddF(base + 1, v.y * w);
    atomAddF(base + 2, v.z * w);
    atomAddF(base + 3, v.w * w);
}

extern "C" void kernel_launch(void* const* d_in, const int* in_sizes, int n_in,
                              void* d_out, int out_size, void* d_ws, size_t ws_size,
                              hipStream_t stream) {
    const float*     x   = (const float*)d_in[0];
    const long long* ei  = (const long long*)d_in[1];  // int64, shape (2, E) row-major
    const float*     W1  = (const float*)d_in[2];
    const float*     b1  = (const float*)d_in[3];
    const float*     W2  = (const float*)d_in[4];
    const float*     b2  = (const float*)d_in[5];
    const float*     W3  = (const float*)d_in[6];
    const float*     b3  = (const float*)d_in[7];
    const float*     Wl  = (const float*)d_in[8];
    const float*     bl  = (const float*)d_in[9];
    float*           out = (float*)d_out;

    const int n = in_sizes[0] / FEAT;  // 100000
    const int e = in_sizes[1] / 2;     // 1600000
    const long long* src = ei;
    const long long* dst = ei + e;

    // workspace: dis[n] | T[n*64] | AGG[n*64]   (~51.6 MB)
    float* dis = (float*)d_ws;
    float* T   = dis + n;
    float* AGG = T + (size_t)n * FEAT;

    const int B = 256;
    const int gN   = (n + B - 1) / B;
    const int gE   = (e + B - 1) / B;
    const int gN16 = (n * 16 + B - 1) / B;
    const int gE16 = (int)(((long long)e * 16 + B - 1) / B);
    const int mt   = n / 16;  // 6250 M-tiles (N divisible by 16)

    // symmetric normalization
    k_init_deg<<<gN, B, 0, stream>>>(dis, n);
    k_accum_deg<<<gE, B, 0, stream>>>(dst, dis, e);
    k_rsqrt_inplace<<<gN, B, 0, stream>>>(dis, n);

    // layer 1: T = x @ W1 ; AGG = b1 + selfloop + edge scatter
    k_gemm_wmma<false, 64, false><<<mt, 128, 0, stream>>>(x, W1, nullptr, T);
    k_init_agg<<<gN16, B, 0, stream>>>(T, dis, b1, AGG, n);
    k_scatter<<<gE16, B, 0, stream>>>(T, dis, src, dst, AGG, e);

    // layer 2: T = relu(AGG) @ W2 ; rebuild AGG
    k_gemm_wmma<true, 64, false><<<mt, 128, 0, stream>>>(AGG, W2, nullptr, T);
    k_init_agg<<<gN16, B, 0, stream>>>(T, dis, b2, AGG, n);
    k_scatter<<<gE16, B, 0, stream>>>(T, dis, src, dst, AGG, e);

    // layer 3
    k_gemm_wmma<true, 64, false><<<mt, 128, 0, stream>>>(AGG, W3, nullptr, T);
    k_init_agg<<<gN16, B, 0, stream>>>(T, dis, b3, AGG, n);
    k_scatter<<<gE16, B, 0, stream>>>(T, dis, src, dst, AGG, e);

    // final linear: out = relu(AGG) @ Wl + bl   (64 -> 8, one wave per 16-row tile)
    k_gemm_wmma<true, 8, true><<<mt, 32, 0, stream>>>(AGG, Wl, bl, out);
}